// KGPLStudent_17652315587461
// MI455X (gfx1250) — compile-verified
//
#include <hip/hip_runtime.h>
#include <math.h>

// KGPL/KGCN forward, B=4096, D=64, K=16, N_ITER=2, gfx1250 wave32.
// One block (8 wave32) per batch element. f32 WMMA (16x16x4) for the 64x64 layers.

#define D 64
#define KNB 16
#define NREL 32

typedef __attribute__((ext_vector_type(2))) float v2f;
typedef __attribute__((ext_vector_type(8))) float v8f;

__global__ __launch_bounds__(256) void kgpl_fwd_kernel(
    const int* __restrict__ user_idx,
    const int* __restrict__ item_idx,
    const float* __restrict__ user_emb,
    const float* __restrict__ ent_emb,
    const float* __restrict__ rel_emb,
    const int* __restrict__ adj_ent,
    const int* __restrict__ adj_rel,
    const float* __restrict__ W0,
    const float* __restrict__ b0,
    const float* __restrict__ W1,
    const float* __restrict__ b1,
    float* __restrict__ out)
{
    __shared__ float sU[D];            // user embedding
    __shared__ float sRel[NREL * D];   // full relation table (8 KB)
    __shared__ float sW[D * D];        // W0 staged (16 KB)
    __shared__ float sA[32 * D];       // matmul A: rows 0-15 hop1, row 16 hop0, 17-31 zero
    __shared__ float sH1[KNB * D];     // layer0 hop1 outputs
    __shared__ float sH0[D];           // layer0 hop0 output
    __shared__ int   sE1[KNB];
    __shared__ int   sR1[KNB];

    const int tid  = threadIdx.x;
    const int lane = tid & 31;
    const int wave = tid >> 5;
    const int l16  = lane & 15;   // 0..15 (neighbor index / tile col / tile row)
    const int hf   = lane >> 4;   // lane half (0/1)
    const int b    = blockIdx.x;

    const int  item  = item_idx[b];
    const long ubase = (long)user_idx[b] * D;

    // ---------------- Phase 1: cooperative staging ----------------
    if (tid < D) sU[tid] = user_emb[ubase + tid];
    for (int i = tid; i < NREL * D; i += 256) sRel[i] = rel_emb[i];
    for (int i = tid; i < D * D; i += 256)    sW[i]   = W0[i];
    if (tid < KNB) {
        sE1[tid] = adj_ent[(long)item * KNB + tid];
        sR1[tid] = adj_rel[(long)item * KNB + tid];
    }
    for (int i = 17 * D + tid; i < 32 * D; i += 256) sA[i] = 0.f;  // zero pad rows
    __syncthreads();

    // ---------------- Phase 2: attention + aggregation, nodes 0..16 ----------------
    // node 0..15 = hop-1 entities (neighbors = hop-2), node 16 = item (neighbors = hop-1)
    for (int node = wave; node < 17; node += 8) {
        int selfId, eid, rid;
        if (node < 16) {
            selfId = sE1[node];
            eid = adj_ent[(long)selfId * KNB + l16];
            rid = adj_rel[(long)selfId * KNB + l16];
        } else {
            selfId = item;
            eid = sE1[l16];
            rid = sR1[l16];
        }
        // score_k = mean_d(u_d * rel[rid]_d); dims split across lane halves
        float p = 0.f;
        {
            const float* rrow = &sRel[rid * D + hf * 32];
            const float* urow = &sU[hf * 32];
            for (int d = 0; d < 32; ++d) p += urow[d] * rrow[d];
        }
        p += __shfl_xor(p, 16, 32);           // both halves now hold full dot
        float s = p * (1.f / 64.f);
        // softmax over the 16 neighbors (duplicated across halves)
        float mx = s;
        for (int off = 8; off >= 1; off >>= 1) mx = fmaxf(mx, __shfl_xor(mx, off, 32));
        float ex = __expf(s - mx);
        float sm = ex;
        for (int off = 8; off >= 1; off >>= 1) sm += __shfl_xor(sm, off, 32);
        const float att = ex / sm;
        // aggregation: lane owns dims (lane) and (lane+32); coalesced 256B row gathers
        float acc0 = 0.f, acc1 = 0.f;
        for (int kk = 0; kk < 16; ++kk) {
            float a  = __shfl(att, kk, 32);
            int   id = __shfl(eid, kk, 32);
            const float* row = &ent_emb[(long)id * D];
            acc0 += a * row[lane];
            acc1 += a * row[lane + 32];
        }
        const float* srow = &ent_emb[(long)selfId * D];
        sA[node * D + lane]      = srow[lane]      + acc0;
        sA[node * D + lane + 32] = srow[lane + 32] + acc1;
    }
    __syncthreads();

    // ---------------- Phase 3: H = relu(A @ W0^T + b0), f32 WMMA 16x16x4 ----------------
    // 8 waves -> 8 output tiles: mt in {0,1}, nt in {0..3}
    {
        const int mt = wave >> 2;
        const int nt = wave & 3;
        v8f c = {};
        for (int kk = 0; kk < 16; ++kk) {
            // ISA A layout (16x4 f32): lanes 0-15 K={0,1}, lanes 16-31 K={2,3}
            const int kb = kk * 4 + hf * 2;
            const int m  = mt * 16 + l16;
            v2f av, bv;
            av.x = sA[m * D + kb];
            av.y = sA[m * D + kb + 1];
            // B(k,n) = W[n][k] (out = x @ W^T), same K striping as A
            const int n = nt * 16 + l16;
            bv.x = sW[n * D + kb];
            bv.y = sW[n * D + kb + 1];
            c = __builtin_amdgcn_wmma_f32_16x16x4_f32(
                    false, av, false, bv, (short)0, c, false, false);
        }
        // C layout: VGPR v -> row v (+8 for upper lane half), col = lane&15
        const int nglob = nt * 16 + l16;
        const float bias = b0[nglob];
        for (int v = 0; v < 8; ++v) {
            const int mloc = v + hf * 8;
            const float h = fmaxf(c[v] + bias, 0.f);
            if (mt == 0)          sH1[mloc * D + nglob] = h;   // hop-1 rows
            else if (mloc == 0)   sH0[nglob] = h;              // hop-0 row (row 16)
        }
    }
    __syncthreads();

    // ---------------- Phase 4: layer 1 hop 0 (1 row) + final dot, wave 0 ----------------
    if (wave == 0) {
        // attention over hop-1 relations again
        float p = 0.f;
        {
            const int rid = sR1[l16];
            const float* rrow = &sRel[rid * D + hf * 32];
            const float* urow = &sU[hf * 32];
            for (int d = 0; d < 32; ++d) p += urow[d] * rrow[d];
        }
        p += __shfl_xor(p, 16, 32);
        float s = p * (1.f / 64.f);
        float mx = s;
        for (int off = 8; off >= 1; off >>= 1) mx = fmaxf(mx, __shfl_xor(mx, off, 32));
        float ex = __expf(s - mx);
        float sm = ex;
        for (int off = 8; off >= 1; off >>= 1) sm += __shfl_xor(sm, off, 32);
        const float att = ex / sm;

        float acc0 = 0.f, acc1 = 0.f;
        for (int kk = 0; kk < 16; ++kk) {
            float a = __shfl(att, kk, 32);
            acc0 += a * sH1[kk * D + lane];
            acc1 += a * sH1[kk * D + lane + 32];
        }
        const float x0 = sH0[lane]      + acc0;   // dim = lane
        const float x1 = sH0[lane + 32] + acc1;   // dim = lane+32

        // out_e = tanh(b1_e + sum_d x_d * W1[e][d]) for e = lane, lane+32
        float o0 = b1[lane], o1 = b1[lane + 32];
        for (int d = 0; d < 32; ++d) {
            const float xd = __shfl(x0, d, 32);
            o0 += xd * W1[lane * D + d];
            o1 += xd * W1[(lane + 32) * D + d];
        }
        for (int d = 0; d < 32; ++d) {
            const float xd = __shfl(x1, d, 32);
            o0 += xd * W1[lane * D + 32 + d];
            o1 += xd * W1[(lane + 32) * D + 32 + d];
        }
        o0 = tanhf(o0);
        o1 = tanhf(o1);

        // final score = <u, item_repr>
        float r = sU[lane] * o0 + sU[lane + 32] * o1;
        for (int off = 16; off >= 1; off >>= 1) r += __shfl_xor(r, off, 32);
        if (lane == 0) out[b] = r;
    }
}

extern "C" void kernel_launch(void* const* d_in, const int* in_sizes, int n_in,
                              void* d_out, int out_size, void* d_ws, size_t ws_size,
                              hipStream_t stream) {
    const int*   user_idx = (const int*)d_in[0];
    const int*   item_idx = (const int*)d_in[1];
    const float* user_emb = (const float*)d_in[2];
    const float* ent_emb  = (const float*)d_in[3];
    const float* rel_emb  = (const float*)d_in[4];
    const int*   adj_ent  = (const int*)d_in[5];
    const int*   adj_rel  = (const int*)d_in[6];
    const float* W0       = (const float*)d_in[7];
    const float* b0       = (const float*)d_in[8];
    const float* W1       = (const float*)d_in[9];
    const float* b1       = (const float*)d_in[10];
    float* out = (float*)d_out;

    const int B = in_sizes[0];
    kgpl_fwd_kernel<<<dim3(B), dim3(256), 0, stream>>>(
        user_idx, item_idx, user_emb, ent_emb, rel_emb,
        adj_ent, adj_rel, W0, b0, W1, b1, out);
}